// SyncRenderingModule_22677427323453
// MI455X (gfx1250) — compile-verified
//
#include <hip/hip_runtime.h>
#include <stdint.h>

typedef __attribute__((ext_vector_type(2))) float v2f;
typedef __attribute__((ext_vector_type(8))) float v8f;

static constexpr int BB = 4;     // batches
static constexpr int NN = 1024;  // notes per batch
static constexpr int PP = 128;   // pitches
static constexpr int SECCH = 8;  // chunks per scan section (8 * 256 = 2048 elems)

#define SRL       137.8f
#define SIGMA     0.49608f            /* 3.6/1000*137.8 */
#define INV_SIG   (1.0f / SIGMA)
#define GWIN      8.0f                /* exp(-2.03*64) underflows to 0.0f */

__device__ __forceinline__ float clip01(float x) {
  return fminf(fmaxf(x, 0.0f), 1.0f);
}

__device__ __forceinline__ int clip_pitch(int p) {
  return p < 0 ? 0 : (p > PP - 1 ? PP - 1 : p);
}

// ---------------------------------------------------------------- zero fill
__global__ void k_zero(float* __restrict__ out, int n4) {
  int i = blockIdx.x * blockDim.x + threadIdx.x;
  int stride = gridDim.x * blockDim.x;
  float4* o = (float4*)out;
  float4 z = make_float4(0.f, 0.f, 0.f, 0.f);
  for (; i < n4; i += stride) o[i] = z;
}

// ------------------------------------------------- pass 1: onset + rect deltas
// One WAVE owns one (b, p): note scalars are wave-uniform loads; the <=17-frame
// Gaussian window is covered by lanes in one parallel step; lane 0 writes the
// rectangle difference-deltas.  A wave writes only its own rows -> race-free
// and deterministic (no atomics).
__global__ void __launch_bounds__(256) k_notes_pre(
    const float* __restrict__ starts, const float* __restrict__ ends,
    const float* __restrict__ vels,   const int*   __restrict__ pitches,
    const unsigned char* __restrict__ valid,
    float* __restrict__ out, int T) {
  int wid  = blockIdx.x * 8 + (threadIdx.x >> 5);   // 0..511
  int lane = threadIdx.x & 31;
  if (wid >= BB * PP) return;
  int b = wid >> 7, p = wid & (PP - 1);
  size_t rowT = (size_t)T;
  float* on  = out + ((size_t)(b * 3 * PP) + p) * rowT;
  float* sus = out + ((size_t)(b * 3 * PP) + PP + p) * rowT;
  float* vel = out + ((size_t)(b * 3 * PP) + 2 * PP + p) * rowT;

  for (int n = 0; n < NN; ++n) {
    int pn = clip_pitch(pitches[b * NN + n]);
    if (pn != p) continue;                 // wave-uniform
    if (!valid[b * NN + n]) continue;
    float s = starts[b * NN + n], e = ends[b * NN + n], v = vels[b * NN + n];
    float sf = s * SRL, ef = e * SRL;

    if (v > 0.0f) {                        // onset Gaussian, lanes parallel
      int t0 = (int)ceilf(sf - GWIN);  if (t0 < 0) t0 = 0;
      int t1 = (int)floorf(sf + GWIN); if (t1 > T - 1) t1 = T - 1;
      int t = t0 + lane;
      if (t <= t1) {
        float d = ((float)t - sf) * INV_SIG;
        on[t] += __expf(-0.5f * d * d) * v;
      }
    }
    if (lane == 0) {                       // rect difference deltas
      float rs = floorf(sf); if (rs < 0.f) rs = 0.f;
      float re = ceilf(ef);  if (re > (float)T) re = (float)T;
      int irs = (int)rs, ire = (int)re;
      if (irs < ire) {
        sus[irs] += 1.0f;
        vel[irs] += v;
        if (ire < T) { sus[ire] -= 1.0f; vel[ire] -= v; }
      }
    }
  }
}

// ------------------------------------------------- pass 2: WMMA prefix scan
// One wave per (row, section); row = (b, channel, pitch), 1024 rows; section =
// 2048 elements.  The wave first derives its scan carry directly from the note
// data (prefix of the delta stream before the section is just
// w*([irs<base]-[ire<base]) summed over matching notes), lane-parallel +
// shfl_xor reduce, so all sections run independently.
// Per 256-chunk: stage to LDS via global_load_async_to_lds_b128, reshape as
// M[k][j] (elem = 16*j + k), D = L @ M via 4 chained V_WMMA_F32_16X16X4_F32
// (L = lower-triangular ones), then a 4-step Hillis-Steele shuffle scan of the
// 16 segment totals.  Stores clipped (all contributions >= 0).
__global__ void __launch_bounds__(256) k_scan(
    const float* __restrict__ starts, const float* __restrict__ ends,
    const float* __restrict__ vels,   const int*   __restrict__ pitches,
    const unsigned char* __restrict__ valid,
    float* __restrict__ out, int T) {
  __shared__ float smem[8 * 256];
  int wave = threadIdx.x >> 5;
  int lane = threadIdx.x & 31;
  int gw = blockIdx.x * 8 + wave;
  int rowid = gw % (BB * 2 * PP);               // 0..1023
  int sec   = gw / (BB * 2 * PP);
  int b = rowid >> 8;
  int within = rowid & 255;
  int ch = within >> 7;                         // 0 sustain, 1 velocity
  int p = within & 127;
  float* R = out + ((size_t)((b * 3 + 1 + ch) * PP) + p) * (size_t)T;

  int totalChunks = (T + 255) / 256;
  int c0 = sec * SECCH;
  int c1 = c0 + SECCH; if (c1 > totalChunks) c1 = totalChunks;
  if (c0 >= totalChunks) return;
  int secBase = c0 * 256;

  // ---- section carry from note data (lane-parallel over notes) ----
  float carry = 0.0f;
  for (int n = lane; n < NN; n += 32) {
    int pn = clip_pitch(pitches[b * NN + n]);
    if (pn == p && valid[b * NN + n]) {
      float sf = starts[b * NN + n] * SRL;
      float ef = ends[b * NN + n] * SRL;
      float rs = floorf(sf); if (rs < 0.f) rs = 0.f;
      float re = ceilf(ef);  if (re > (float)T) re = (float)T;
      int irs = (int)rs, ire = (int)re;
      if (irs < ire) {
        float w = ch ? vels[b * NN + n] : 1.0f;
        if (irs < secBase) carry += w;
        if (ire < secBase) carry -= w;          // ire<secBase<=T implies ire<T
      }
    }
  }
#pragma unroll
  for (int d = 16; d >= 1; d >>= 1) carry += __shfl_xor(carry, d, 32);
  // carry is now wave-uniform

  int j   = lane & 15;
  int hi  = lane >> 4;
  int off = hi ? 2 : 0;

  // A = slices of lower-triangular ones, per 16x4 f32 A-matrix VGPR layout.
  float a0[4], a1[4];
#pragma unroll
  for (int k = 0; k < 4; ++k) {
    int kb = 4 * k + off;
    a0[k] = (kb     <= j) ? 1.0f : 0.0f;
    a1[k] = (kb + 1 <= j) ? 1.0f : 0.0f;
  }

  uint32_t ldsBase = (uint32_t)(uintptr_t)(&smem[0]) + (uint32_t)(wave * 1024);
  float* wsmem = &smem[wave * 256];
  bool aligned8 = ((T & 7) == 0);

  for (int c = c0; c < c1; ++c) {
    int base = c * 256;
    bool full = (base + 256 <= T) && aligned8;

    float b0[4], b1[4];
    if (full) {
      // ---- CDNA5 async global -> LDS staging (2 x b128 per lane = 1 KB) ----
      uint32_t l0 = ldsBase + (uint32_t)(lane * 16);
      uint32_t l1 = l0 + 512u;
      uint32_t v0o = (uint32_t)(lane * 16);
      uint32_t v1o = v0o + 512u;
      const float* src = R + base;              // wave-uniform
      asm volatile("global_load_async_to_lds_b128 %0, %1, %2"
                   :: "v"(l0), "v"(v0o), "s"(src) : "memory");
      asm volatile("global_load_async_to_lds_b128 %0, %1, %2"
                   :: "v"(l1), "v"(v1o), "s"(src) : "memory");
      asm volatile("s_wait_asynccnt 0x0" ::: "memory");
#pragma unroll
      for (int k = 0; k < 4; ++k) {
        int e = 16 * j + 4 * k + off;           // B[k'][j] = M[k'][j]
        b0[k] = wsmem[e];
        b1[k] = wsmem[e + 1];
      }
    } else {
#pragma unroll
      for (int k = 0; k < 4; ++k) {
        int e = base + 16 * j + 4 * k + off;
        b0[k] = (e < T)     ? R[e]     : 0.0f;
        b1[k] = (e + 1 < T) ? R[e + 1] : 0.0f;
      }
    }

    v8f acc = {0.f, 0.f, 0.f, 0.f, 0.f, 0.f, 0.f, 0.f};
#pragma unroll
    for (int k = 0; k < 4; ++k) {
      v2f A = {a0[k], a1[k]};
      v2f Bm = {b0[k], b1[k]};
      acc = __builtin_amdgcn_wmma_f32_16x16x4_f32(
          false, A, false, Bm, (short)0, acc, false, false);
    }

    // Segment totals S[j] = D[15][j] live in acc[7] of lanes 16..31.
    // Fetch own total, 4-step Hillis-Steele scan within 16-lane groups.
    float d7 = acc[7];
    float sOwn = __shfl(d7, 16 + j, 32);
    float incl = sOwn;
#pragma unroll
    for (int dlt = 1; dlt < 16; dlt <<= 1) {
      float tshf = __shfl_up(incl, dlt, 16);
      if (j >= dlt) incl += tshf;
    }
    float excl = incl - sOwn;
    float tot  = __shfl(incl, 15, 32);          // uniform chunk total
    float addv = excl + carry;
    carry += tot;

    int outBase = base + 16 * j + hi * 8;       // element i = r + hi*8
    if (full) {
      float4 st0 = make_float4(clip01(acc[0] + addv), clip01(acc[1] + addv),
                               clip01(acc[2] + addv), clip01(acc[3] + addv));
      float4 st1 = make_float4(clip01(acc[4] + addv), clip01(acc[5] + addv),
                               clip01(acc[6] + addv), clip01(acc[7] + addv));
      *(float4*)(R + outBase)     = st0;
      *(float4*)(R + outBase + 4) = st1;
    } else {
#pragma unroll
      for (int r = 0; r < 8; ++r) {
        int idx = outBase + r;
        if (idx < T) R[idx] = clip01(acc[r] + addv);
      }
    }
  }
}

// ------------------------------------------------- pass 3: release tails
// One wave per (b, p); added AFTER the scan so the tails are not integrated.
__global__ void __launch_bounds__(256) k_notes_post(
    const float* __restrict__ starts, const float* __restrict__ ends,
    const float* __restrict__ vels,   const int*   __restrict__ pitches,
    const unsigned char* __restrict__ valid,
    float* __restrict__ out, int T) {
  int wid  = blockIdx.x * 8 + (threadIdx.x >> 5);
  int lane = threadIdx.x & 31;
  if (wid >= BB * PP) return;
  int b = wid >> 7, p = wid & (PP - 1);
  size_t rowT = (size_t)T;
  float* sus = out + ((size_t)(b * 3 * PP) + PP + p) * rowT;
  float* vel = out + ((size_t)(b * 3 * PP) + 2 * PP + p) * rowT;

  for (int n = 0; n < NN; ++n) {
    int pn = clip_pitch(pitches[b * NN + n]);
    if (pn != p) continue;
    if (!valid[b * NN + n]) continue;
    float e = ends[b * NN + n], v = vels[b * NN + n];
    float ef = e * SRL;
    int t0 = (int)ceilf(ef);          if (t0 < 0) t0 = 0;   // rel: t >= ef
    int t1 = (int)floorf(ef + GWIN);  if (t1 > T - 1) t1 = T - 1;
    int t = t0 + lane;                                      // window <= 9
    if (t <= t1) {
      float d = ((float)t - ef) * INV_SIG;
      float g = __expf(-0.5f * d * d) * 0.1f;
      sus[t] = fminf(sus[t] + g, 1.0f);
      vel[t] = fminf(vel[t] + g * v, 1.0f);
    }
  }
}

// ------------------------------------------------- pass 4: clip onset channel
__global__ void k_clip_onset(float* __restrict__ out, int PT4, int n4) {
  int i = blockIdx.x * blockDim.x + threadIdx.x;
  int stride = gridDim.x * blockDim.x;
  for (; i < n4; i += stride) {
    int b = i / PT4;
    int off = i - b * PT4;
    float4* ptr = (float4*)out + (size_t)b * 3 * PT4 + off;
    float4 x = *ptr;
    x.x = clip01(x.x); x.y = clip01(x.y); x.z = clip01(x.z); x.w = clip01(x.w);
    *ptr = x;
  }
}

extern "C" void kernel_launch(void* const* d_in, const int* in_sizes, int n_in,
                              void* d_out, int out_size, void* d_ws, size_t ws_size,
                              hipStream_t stream) {
  (void)in_sizes; (void)n_in; (void)d_ws; (void)ws_size;
  const float* starts        = (const float*)d_in[0];
  const float* ends          = (const float*)d_in[1];
  const float* vels          = (const float*)d_in[2];
  const int*   pitches       = (const int*)d_in[3];
  const unsigned char* valid = (const unsigned char*)d_in[4];
  float* out = (float*)d_out;

  const int T  = out_size / (BB * 3 * PP);   // 16536 for dur=120
  const int n4 = out_size / 4;

  k_zero<<<(n4 + 255) / 256, 256, 0, stream>>>(out, n4);

  // 512 waves, 8 waves per 256-thread block
  k_notes_pre<<<(BB * PP) / 8, 256, 0, stream>>>(
      starts, ends, vels, pitches, valid, out, T);

  // 1024 rows x nSec sections, 8 waves per block
  const int nSec = (T + SECCH * 256 - 1) / (SECCH * 256);
  k_scan<<<((BB * 2 * PP) / 8) * nSec, 256, 0, stream>>>(
      starts, ends, vels, pitches, valid, out, T);

  k_notes_post<<<(BB * PP) / 8, 256, 0, stream>>>(
      starts, ends, vels, pitches, valid, out, T);

  const int PT4 = (PP * T) / 4;
  const int on4 = (BB * PP * T) / 4;
  k_clip_onset<<<(on4 + 255) / 256, 256, 0, stream>>>(out, PT4, on4);
}